// Yolo_Loss_28269474742964
// MI455X (gfx1250) — compile-verified
//
#include <hip/hip_runtime.h>
#include <hip/hip_bf16.h>
#include <math.h>

// ---------------------------------------------------------------------------
// YOLO loss for MI455X (gfx1250, wave32).
// Strategy: only channels 0..4 are needed everywhere; the 80 class channels
// are loaded only by waves that contain a positive anchor cell. Streamed,
// non-temporal, coalesced loads; labels staged to LDS via the CDNA5 async
// global->LDS path; block reduction finished with an exact f32 WMMA
// (V_WMMA_F32_16X16X4_F32, B = ones). Deterministic: per-block partial sums
// in workspace, summed in fixed order by a single finalize block.
// ---------------------------------------------------------------------------

#define YL_T 20           // targets per batch
#define YL_LAB (YL_T * 5) // floats per batch in label tensor

typedef __attribute__((ext_vector_type(2))) float v2f;
typedef __attribute__((ext_vector_type(8))) float v8f;

#if defined(__gfx1250__) && __has_builtin(__builtin_amdgcn_global_load_async_to_lds_b32) && __has_builtin(__builtin_amdgcn_s_wait_asynccnt)
#define YL_ASYNC 1
#endif
#if defined(__gfx1250__) && __has_builtin(__builtin_amdgcn_wmma_f32_16x16x4_f32)
#define YL_WMMA 1
#endif

__device__ __forceinline__ float yl_sigm(float x) { return 1.0f / (1.0f + __expf(-x)); }

__device__ __forceinline__ float yl_iou(float ax, float ay, float aw, float ah,
                                        float bx, float by, float bw, float bh) {
  float A1x = ax - aw * 0.5f, A1y = ay - ah * 0.5f;
  float A2x = ax + aw * 0.5f, A2y = ay + ah * 0.5f;
  float B1x = bx - bw * 0.5f, B1y = by - bh * 0.5f;
  float B2x = bx + bw * 0.5f, B2y = by + bh * 0.5f;
  float ix = fmaxf(fminf(A2x, B2x) - fmaxf(A1x, B1x), 0.0f);
  float iy = fmaxf(fminf(A2y, B2y) - fmaxf(A1y, B1y), 0.0f);
  float ai = ix * iy;
  float au = aw * ah + bw * bh - ai;
  return ai / (au + 1e-7f);
}

__device__ __forceinline__ float yl_diou(float ax, float ay, float aw, float ah,
                                         float bx, float by, float bw, float bh) {
  float A1x = ax - aw * 0.5f, A1y = ay - ah * 0.5f;
  float A2x = ax + aw * 0.5f, A2y = ay + ah * 0.5f;
  float B1x = bx - bw * 0.5f, B1y = by - bh * 0.5f;
  float B2x = bx + bw * 0.5f, B2y = by + bh * 0.5f;
  float ix = fmaxf(fminf(A2x, B2x) - fmaxf(A1x, B1x), 0.0f);
  float iy = fmaxf(fminf(A2y, B2y) - fmaxf(A1y, B1y), 0.0f);
  float ai = ix * iy;
  float au = aw * ah + bw * bh - ai;
  float iou = ai / (au + 1e-7f);
  float ox = fmaxf(fmaxf(A2x, B2x) - fminf(A1x, B1x), 0.0f);
  float oy = fmaxf(fmaxf(A2y, B2y) - fminf(A1y, B1y), 0.0f);
  float r2 = (ax - bx) * (ax - bx) + (ay - by) * (ay - by);
  float c2 = ox * ox + oy * oy;
  return iou - r2 / (c2 + 1e-7f);
}

__device__ __forceinline__ float yl_wred(float v) {
#pragma unroll
  for (int m = 16; m >= 1; m >>= 1) v += __shfl_xor(v, m, 32);
  return v;
}

// One thread per anchor cell (b,i,j,a). Per-block partial sums of
// {npos, nneg, sum_box, sum_obj_pos, sum_obj_neg, sum_cls} -> outBase[slot*6..]
__global__ __launch_bounds__(256) void yolo_scale_kernel(
    const float* __restrict__ raw, const float* __restrict__ label,
    float* __restrict__ outBase, int S, float stride,
    float aw0, float ah0, float aw1, float ah1, float aw2, float ah2) {
  const int SS = S * S;
  const int b = blockIdx.y / 3;
  const int a = blockIdx.y % 3;
  const float aw = (a == 0) ? aw0 : ((a == 1) ? aw1 : aw2);
  const float ah = (a == 0) ? ah0 : ((a == 1) ? ah1 : ah2);

  __shared__ float labL[YL_LAB];
  __shared__ float sw[8][16];  // per-wave partials, padded to 16 for WMMA A rows

  if (threadIdx.x < 128) ((float*)sw)[threadIdx.x] = 0.0f;

#ifdef YL_ASYNC
  if (threadIdx.x < YL_LAB) {
    __builtin_amdgcn_global_load_async_to_lds_b32(
        (__attribute__((address_space(1))) int*)(label + (size_t)b * YL_LAB + threadIdx.x),
        (__attribute__((address_space(3))) int*)(&labL[threadIdx.x]),
        0, 0);
  }
  __builtin_amdgcn_s_wait_asynccnt(0);
#else
  if (threadIdx.x < YL_LAB) labL[threadIdx.x] = label[(size_t)b * YL_LAB + threadIdx.x];
#endif
  __syncthreads();

  const int p = blockIdx.x * blockDim.x + threadIdx.x;
  const bool valid = (p < SS);
  const int pc = valid ? p : (SS - 1);
  const int i = pc / S;
  const int j = pc - i * S;

  const float* base = raw + (size_t)(b * 3 + a) * 85 * (size_t)SS;

  // box + objectness channels (streamed once -> non-temporal)
  float tx = __builtin_nontemporal_load(base + pc);
  float ty = __builtin_nontemporal_load(base + (size_t)SS + pc);
  float tw = __builtin_nontemporal_load(base + 2 * (size_t)SS + pc);
  float th = __builtin_nontemporal_load(base + 3 * (size_t)SS + pc);
  float tb = __builtin_nontemporal_load(base + 4 * (size_t)SS + pc);

  float px = (yl_sigm(tx) * 1.05f - 0.025f + (float)j) * stride;
  float py = (yl_sigm(ty) * 1.05f - 0.025f + (float)i) * stride;
  float pw = __expf(tw) * aw;
  float ph = __expf(th) * ah;
  const float invS = 1.0f / stride;  // stride is a power of 2 -> exact

  float maxiou = 0.0f;
  int tpos = -1;
#pragma unroll
  for (int t = 0; t < YL_T; ++t) {
    float lcls = labL[t * 5 + 0];
    float lx = labL[t * 5 + 1], ly = labL[t * 5 + 2];
    float lw = labL[t * 5 + 3], lh = labL[t * 5 + 4];
    maxiou = fmaxf(maxiou, yl_iou(px, py, pw, ph, lx, ly, lw, lh));
    // reference quirk: ii = floor(label[...,0]/stride) (class!), jj = floor(x/stride)
    int jjt = (int)floorf(lx * invS);    // indexes first spatial dim (i)
    int iit = (int)floorf(lcls * invS);  // indexes second spatial dim (j)
    float ain = fminf(lw, aw) * fminf(lh, ah);
    float aio = ain / (lw * lh + aw * ah - ain + 1e-7f);
    if (i == jjt && j == iit && aio > 0.2f) tpos = t;  // last match wins (scan order)
  }

  const bool pos = valid && (tpos >= 0);
  const bool neg = valid && (maxiou < 0.7f);

  float pobj = yl_sigm(tb);
  float lp = fmaxf(__logf(pobj), -100.0f);
  float lq = fmaxf(__logf(1.0f - pobj), -100.0f);
  float bco = pos ? -lp : -lq;  // bce(pobj, target_obj) with target_obj = pos?1:0

  float r0 = pos ? 1.0f : 0.0f;
  float r1 = neg ? 1.0f : 0.0f;
  float r2 = 0.0f;
  float r3 = pos ? bco : 0.0f;
  float r4 = neg ? bco : 0.0f;
  float r5 = 0.0f;

  int cpos = -1;
  if (pos) {
    float lx = labL[tpos * 5 + 1], ly = labL[tpos * 5 + 2];
    float lw = labL[tpos * 5 + 3], lh = labL[tpos * 5 + 4];
    r2 = 1.0f - yl_diou(px, py, pw, ph, lx, ly, lw, lh);
    cpos = (int)labL[tpos * 5 + 0];
  }

  // Class BCE only matters at positive cells: skip 80/85 of the bytes for
  // waves with no positive lane. Loads stay coalesced within the wave.
  if (__any(pos)) {
    float scls = 0.0f;
#pragma unroll 4
    for (int k = 0; k < 80; ++k) {
      if ((k & 15) == 0)
        __builtin_prefetch(base + (size_t)(5 + k + 16) * SS + pc, 0, 0);
      float lg = __builtin_nontemporal_load(base + (size_t)(5 + k) * SS + pc);
      float qk = yl_sigm(lg);
      float lpk = fmaxf(__logf(qk), -100.0f);
      float lqk = fmaxf(__logf(1.0f - qk), -100.0f);
      scls += (k == cpos) ? -lpk : -lqk;
    }
    r5 = pos ? scls : 0.0f;
  }

  // wave32 reduction (exact, fixed order)
  r0 = yl_wred(r0); r1 = yl_wred(r1); r2 = yl_wred(r2);
  r3 = yl_wred(r3); r4 = yl_wred(r4); r5 = yl_wred(r5);

  const int w = threadIdx.x >> 5;
  const int lane = threadIdx.x & 31;
  float mine = (lane == 0) ? r0 : (lane == 1) ? r1 : (lane == 2) ? r2
             : (lane == 3) ? r3 : (lane == 4) ? r4 : r5;
  if (lane < 6) sw[w][lane] = mine;
  __syncthreads();

  if (threadIdx.x < 32) {  // wave 0, full EXEC (WMMA requirement)
#ifdef YL_WMMA
    // A[m,k] = sw[k][m] (16x4 f32), B = ones (4x16) -> D[m,n] = sum_k sw[k][m].
    // Two chained WMMAs accumulate waves 0..3 then 4..7. Exact f32 math.
    v2f A1, A2, Bm;
    if (lane < 16) {
      A1.x = sw[0][lane];      A1.y = sw[1][lane];
      A2.x = sw[4][lane];      A2.y = sw[5][lane];
    } else {
      A1.x = sw[2][lane - 16]; A1.y = sw[3][lane - 16];
      A2.x = sw[6][lane - 16]; A2.y = sw[7][lane - 16];
    }
    Bm.x = 1.0f; Bm.y = 1.0f;
    v8f C = {};
    C = __builtin_amdgcn_wmma_f32_16x16x4_f32(false, A1, false, Bm, (short)0, C, false, false);
    C = __builtin_amdgcn_wmma_f32_16x16x4_f32(false, A2, false, Bm, (short)0, C, false, false);
    if (lane == 0) {
      float* o = outBase + (size_t)(blockIdx.y * gridDim.x + blockIdx.x) * 6;
      o[0] = C[0]; o[1] = C[1]; o[2] = C[2];
      o[3] = C[3]; o[4] = C[4]; o[5] = C[5];
    }
#else
    if (lane == 0) {
      float* o = outBase + (size_t)(blockIdx.y * gridDim.x + blockIdx.x) * 6;
#pragma unroll
      for (int q = 0; q < 6; ++q) {
        float s = 0.0f;
#pragma unroll
        for (int ww = 0; ww < 8; ++ww) s += sw[ww][q];
        o[q] = s;
      }
    }
#endif
  }
}

// Deterministic final reduction over per-block slots (fixed order), then the
// per-scale normalization + active gating, then d_out = {total, lb, lo, lc}.
__global__ __launch_bounds__(256) void yolo_finalize_kernel(
    const float* __restrict__ ws, int nb0, int nb1, int nb2, float Bf,
    float* __restrict__ out) {
  __shared__ float red[256][6];
  const int tid = threadIdx.x;
  float lb = 0.0f, lo = 0.0f, lc = 0.0f;
  int nbs[3] = {nb0, nb1, nb2};
  size_t off = 0;
  for (int s = 0; s < 3; ++s) {
    int nb = nbs[s];
    float acc[6] = {0.f, 0.f, 0.f, 0.f, 0.f, 0.f};
    for (int slot = tid; slot < nb; slot += 256) {
      const float* p = ws + off + (size_t)slot * 6;
#pragma unroll
      for (int q = 0; q < 6; ++q) acc[q] += p[q];
    }
#pragma unroll
    for (int q = 0; q < 6; ++q) red[tid][q] = acc[q];
    __syncthreads();
    for (int h = 128; h > 0; h >>= 1) {
      if (tid < h) {
#pragma unroll
        for (int q = 0; q < 6; ++q) red[tid][q] += red[tid + h][q];
      }
      __syncthreads();
    }
    if (tid == 0) {
      float npos = red[0][0], nneg = red[0][1];
      float sbox = red[0][2], sop = red[0][3], son = red[0][4], scls = red[0][5];
      float dnp = fmaxf(npos, 1.0f), dnn = fmaxf(nneg, 1.0f);
      float l_box = sbox / dnp / Bf;
      float l_obj = sop / dnp / Bf + son / dnn / Bf;
      float l_cls = scls / (dnp * 80.0f) / Bf;
      float act = (npos > 0.0f && nneg > 0.0f) ? 1.0f : 0.0f;
      lb += l_box * act;
      lo += l_obj * act;
      lc += l_cls * act;
    }
    __syncthreads();
    off += (size_t)nb * 6;
  }
  if (tid == 0) {
    out[0] = lb + lo + lc;
    out[1] = lb;
    out[2] = lo;
    out[3] = lc;
  }
}

extern "C" void kernel_launch(void* const* d_in, const int* in_sizes, int n_in,
                              void* d_out, int out_size, void* d_ws, size_t ws_size,
                              hipStream_t stream) {
  (void)n_in; (void)out_size; (void)ws_size;
  const float* p76 = (const float*)d_in[0];
  const float* p38 = (const float*)d_in[1];
  const float* p19 = (const float*)d_in[2];
  const float* label = (const float*)d_in[3];
  const int B = in_sizes[3] / YL_LAB;  // label is (B, 20, 5)
  float* ws = (float*)d_ws;

  struct Sc { const float* raw; int S; float stride; float an[6]; };
  const Sc scs[3] = {
      {p76, 76, 8.0f,  {28.f, 28.f, 46.f, 45.f, 64.f, 66.f}},
      {p38, 38, 16.0f, {102.f, 74.f, 78.f, 115.f, 132.f, 113.f}},
      {p19, 19, 32.0f, {149.f, 163.f, 174.f, 268.f, 257.f, 176.f}},
  };

  int nb[3];
  size_t off = 0;
  for (int s = 0; s < 3; ++s) {
    const int SS = scs[s].S * scs[s].S;
    const int gx = (SS + 255) / 256;
    nb[s] = gx * B * 3;
    dim3 grid(gx, B * 3);
    yolo_scale_kernel<<<grid, 256, 0, stream>>>(
        scs[s].raw, label, ws + off, scs[s].S, scs[s].stride,
        scs[s].an[0], scs[s].an[1], scs[s].an[2],
        scs[s].an[3], scs[s].an[4], scs[s].an[5]);
    off += (size_t)nb[s] * 6;
  }
  yolo_finalize_kernel<<<1, 256, 0, stream>>>(ws, nb[0], nb[1], nb[2],
                                              (float)B, (float*)d_out);
}